// BiSTGNNv7_63393717289320
// MI455X (gfx1250) — compile-verified
//
#include <hip/hip_runtime.h>
#include <math.h>

// BiSTGNN forward for MI455X (gfx1250, wave32).
// Heavy GEMMs (graph attention A=tanh(relu(X X^T)), GCN A@X@W, tr-GRU
// recurrent h@Whh^T) run on V_WMMA_F32_16X16X4_F32 (fp32-exact).
// GCN LDS tiles are fed by the Tensor Data Mover (tensor_load_to_lds +
// s_wait_tensorcnt). A (140MB) is never materialized: fused tile recompute.

typedef __attribute__((ext_vector_type(2))) float v2f;
typedef __attribute__((ext_vector_type(8))) float v8f;
typedef __attribute__((ext_vector_type(4))) unsigned int u32x4;
typedef __attribute__((ext_vector_type(8))) int i32x8;
typedef __attribute__((ext_vector_type(4))) int i32x4;

#define B_    8
#define T_    96
#define N_    2000
#define L_    32
#define E_    32
#define DT_   4
#define ROWS_ 2096   // N_+T_
#define RT_   131    // ROWS_/16
#define K1_   128    // T_+E_
#define KT_   2036   // N_+DT_+E_
#define G3_   6000   // 3*N_

__device__ __forceinline__ float elu_(float x){ return x > 0.f ? x : (expf(x) - 1.f); }
__device__ __forceinline__ float sig_(float x){ return 1.f / (1.f + expf(-x)); }

__device__ __forceinline__ v8f vzero8(){
  v8f v;
  #pragma unroll
  for (int i = 0; i < 8; ++i) v[i] = 0.f;
  return v;
}
__device__ __forceinline__ v2f vzero2(){ v2f v; v.x = 0.f; v.y = 0.f; return v; }

__device__ __forceinline__ v8f wmma4(v2f a, v2f b, v8f c){
  // D = A(16x4) * B(4x16) + C, fp32
  return __builtin_amdgcn_wmma_f32_16x16x4_f32(false, a, false, b, (short)0, c, false, false);
}

// TDM: 1-row tensor descriptor copying nelem fp32 global -> LDS (TENSORcnt).
__device__ __forceinline__ void tdm_load_1d(unsigned int lds_off, const float* gsrc, int nelem)
{
  unsigned long long ga = (unsigned long long)(size_t)gsrc;
  u32x4 g0;
  g0[0] = 1u;                                   // count=1, user mode, no gather
  g0[1] = lds_off;                              // LDS byte address
  g0[2] = (unsigned int)ga;                     // global_addr[31:0]
  g0[3] = (unsigned int)(ga >> 32) | (2u << 30);// global_addr[56:32] | type=2
  i32x8 g1;
  g1[0] = 0x00020000;                           // data_size=2 (4 bytes)
  g1[1] = (nelem & 0xffff) << 16;               // tensor_dim0[15:0]
  g1[2] = (nelem >> 16) | (1 << 16);            // tensor_dim0[31:16] | tensor_dim1=1
  g1[3] = (nelem & 0xffff) << 16;               // tile_dim0
  g1[4] = 1;                                    // tile_dim1=1, tile_dim2=0
  g1[5] = nelem;                                // tensor_dim0_stride[31:0]
  g1[6] = 0;                                    // stride hi, dim1_stride lo
  g1[7] = 0;
  i32x4 gz4; gz4[0] = 0; gz4[1] = 0; gz4[2] = 0; gz4[3] = 0;
  i32x8 gz8;
  #pragma unroll
  for (int i = 0; i < 8; ++i) gz8[i] = 0;
  // 6-arg form (clang-23 / therock-10.0 headers on this toolchain)
  __builtin_amdgcn_tensor_load_to_lds(g0, g1, gz4, gz4, gz8, 0);
}

// ---- 1) spatial MLP: Xs = elu([x^T | sp_emb] W1 + b1) W2 + b2 -> X rows [0,N) ----
__global__ void k_spatial(const float* __restrict__ x, const float* __restrict__ sp_emb,
                          const float* __restrict__ W1, const float* __restrict__ b1,
                          const float* __restrict__ W2, const float* __restrict__ b2,
                          float* __restrict__ X)
{
  __shared__ float hb[8][32];
  int wave = threadIdx.x >> 5, lane = threadIdx.x & 31;
  int row = blockIdx.x * 8 + wave;            // 16000 = B_*N_
  int b = row / N_, n = row % N_;
  float acc = b1[lane];
  for (int k = 0; k < K1_; ++k) {
    float v = (k < T_) ? x[(b*T_ + k)*N_ + n] : sp_emb[n*E_ + (k - T_)];
    acc = fmaf(v, W1[k*L_ + lane], acc);
  }
  hb[wave][lane] = elu_(acc);
  float o = b2[lane];
  #pragma unroll
  for (int l = 0; l < L_; ++l) o = fmaf(hb[wave][l], W2[l*L_ + lane], o);
  X[(b*ROWS_ + n)*L_ + lane] = o;
}

// ---- 2) temporal GRU input gates: GI = xt @ Wih^T + bih  (B,T,96) ----
__global__ void k_tgi(const float* __restrict__ x, const float* __restrict__ x_mark,
                      const float* __restrict__ t_emb, const float* __restrict__ Wih,
                      const float* __restrict__ bih, float* __restrict__ GI)
{
  int wave = threadIdx.x >> 5, lane = threadIdx.x & 31;
  int row = blockIdx.x * 8 + wave;            // 768 = B_*T_
  int b = row / T_, t = row % T_;
  float a0 = bih[lane], a1 = bih[lane+32], a2 = bih[lane+64];
  const float* xr = x + (size_t)(b*T_ + t)*N_;
  for (int k = 0; k < KT_; ++k) {
    float v = (k < N_) ? xr[k]
            : (k < N_+DT_ ? x_mark[(b*T_ + t)*DT_ + (k - N_)]
                          : t_emb[t*E_ + (k - N_ - DT_)]);
    a0 = fmaf(v, Wih[(size_t)(lane    )*KT_ + k], a0);
    a1 = fmaf(v, Wih[(size_t)(lane+32)*KT_ + k], a1);
    a2 = fmaf(v, Wih[(size_t)(lane+64)*KT_ + k], a2);
  }
  float* g = GI + (b*T_ + t)*96;
  g[lane] = a0; g[lane+32] = a1; g[lane+64] = a2;
}

// ---- 3) temporal GRU scan (L=32, tiny; one wave per batch) -> X rows [N, N+T) ----
__global__ void k_tscan(const float* __restrict__ GI, const float* __restrict__ Whh,
                        const float* __restrict__ bhh, float* __restrict__ X)
{
  int b = threadIdx.x >> 5, lane = threadIdx.x & 31;
  float h = 0.f;
  for (int t = 0; t < T_; ++t) {
    float gr = bhh[lane], gz = bhh[lane+32], gn = bhh[lane+64];
    #pragma unroll
    for (int l = 0; l < L_; ++l) {
      float hv = __shfl(h, l, 32);
      gr = fmaf(hv, Whh[(lane    )*L_ + l], gr);
      gz = fmaf(hv, Whh[(lane+32)*L_ + l], gz);
      gn = fmaf(hv, Whh[(lane+64)*L_ + l], gn);
    }
    const float* g = GI + (b*T_ + t)*96;
    float r = sig_(g[lane] + gr);
    float z = sig_(g[lane+32] + gz);
    float n = tanhf(g[lane+64] + r*gn);
    h = (1.f - z)*n + z*h;
    X[(b*ROWS_ + N_ + t)*L_ + lane] = h;
  }
}

// ---- 4) fused TN/GCN layer: Xout = elu( tanh(relu(Xref Xref^T)) @ Xin @ W + b ) ----
__global__ void __launch_bounds__(256) k_gcn(const float* __restrict__ Xref,
                                             const float* __restrict__ Xin,
                                             float* __restrict__ Xout,
                                             const float* __restrict__ W,
                                             const float* __restrict__ bias)
{
  // unified LDS block: [0,512)=Xref m-tile, [512,1024)=Xin m-tile,
  // [1024,3072)=per-wave 16x16 A tiles, [3072,7168)=per-wave 16x32 accumulators
  __shared__ float smem[7168];
  float* sXm  = smem;
  float* sXi  = smem + 512;
  int wave = threadIdx.x >> 5, lane = threadIdx.x & 31;
  float* sSw  = smem + 1024 + wave*256;
  float* sAw  = smem + 3072 + wave*512;
  unsigned int ldsBase = (unsigned int)(size_t)(void*)smem;   // LDS byte offset
  int b = blockIdx.y;
  int rtile = blockIdx.x * 8 + wave;
  bool act = rtile < RT_;
  int m16 = lane & 15;
  int off = (lane >> 4) << 1;           // K lo/hi half per half-wave
  int mmHi = (lane >> 4) << 3;

  v2f aA[8];                            // this wave's 16x32 Xref rows as A-frags
  v2f wB[8][2];                         // 32x32 gcn weight as B-frags
  if (act) {
    const float* xr = Xref + (size_t)(b*ROWS_ + rtile*16 + m16)*L_;
    #pragma unroll
    for (int kc = 0; kc < 8; ++kc) aA[kc] = *(const v2f*)(xr + kc*4 + off);
  }
  #pragma unroll
  for (int kc = 0; kc < 8; ++kc)
    #pragma unroll
    for (int nh = 0; nh < 2; ++nh) {
      int n = nh*16 + m16, ks = kc*4 + off;
      wB[kc][nh].x = W[ks*L_ + n]; wB[kc][nh].y = W[(ks+1)*L_ + n];
    }

  v8f acc0 = vzero8(), acc1 = vzero8();
  for (int m = 0; m < RT_; ++m) {
    __syncthreads();                    // previous-iteration readers done
    if (wave == 0) {
      // async DMA both 16x32 tiles into LDS via Tensor Data Mover
      tdm_load_1d(ldsBase,           Xref + (size_t)(b*ROWS_ + m*16)*L_, 512);
      tdm_load_1d(ldsBase + 512*4,   Xin  + (size_t)(b*ROWS_ + m*16)*L_, 512);
      __builtin_amdgcn_s_wait_tensorcnt(0);
    }
    if (m + 1 < RT_) __builtin_prefetch(Xin + (size_t)(b*ROWS_ + (m+1)*16)*L_, 0, 1);
    __syncthreads();                    // tiles visible to all waves
    if (act) {
      // S = Xr(16x32) . Xm(16x32)^T : B-frag of Xm^T == A-frag pattern of Xm
      v8f c = vzero8();
      #pragma unroll
      for (int kc = 0; kc < 8; ++kc) {
        v2f bx = *(const v2f*)(sXm + m16*L_ + kc*4 + off);
        c = wmma4(aA[kc], bx, c);
      }
      // A = tanh(relu(S)); redistribute D-layout -> row-major via LDS
      #pragma unroll
      for (int i = 0; i < 8; ++i) {
        float s = c[i];
        sSw[(i + mmHi)*16 + m16] = (s > 0.f) ? tanhf(s) : 0.f;
      }
      // acc += A(16x16) @ Xin_m(16x32)
      #pragma unroll
      for (int kc = 0; kc < 4; ++kc) {
        int ks = kc*4 + off;
        v2f a2 = *(const v2f*)(sSw + m16*16 + ks);
        v2f b0, b1;
        b0.x = sXi[ks*L_ + m16];      b0.y = sXi[(ks+1)*L_ + m16];
        b1.x = sXi[ks*L_ + 16 + m16]; b1.y = sXi[(ks+1)*L_ + 16 + m16];
        acc0 = wmma4(a2, b0, acc0);
        acc1 = wmma4(a2, b1, acc1);
      }
    }
  }
  // Y = acc(16x32) @ W(32x32) + bias, elu
  if (act) {
    #pragma unroll
    for (int i = 0; i < 8; ++i) {
      sAw[(i + mmHi)*32 + m16]      = acc0[i];
      sAw[(i + mmHi)*32 + 16 + m16] = acc1[i];
    }
    #pragma unroll
    for (int nh = 0; nh < 2; ++nh) {
      v8f yc = vzero8();
      #pragma unroll
      for (int kc = 0; kc < 8; ++kc) {
        int ks = kc*4 + off;
        v2f a3 = *(const v2f*)(sAw + m16*32 + ks);
        yc = wmma4(a3, wB[kc][nh], yc);
      }
      int d = nh*16 + m16;
      float bv = bias[d];
      #pragma unroll
      for (int i = 0; i < 8; ++i)
        Xout[(size_t)(b*ROWS_ + rtile*16 + (i + mmHi))*L_ + d] = elu_(yc[i] + bv);
    }
  }
}

// ---- 5) fr MLP: combined = elu(Xs2 W1 + b1) W2 + b2  (B,N,T layout) ----
__global__ void k_fr(const float* __restrict__ X, const float* __restrict__ W1,
                     const float* __restrict__ b1, const float* __restrict__ W2,
                     const float* __restrict__ b2, float* __restrict__ comb)
{
  __shared__ float hb[8][96];
  int wave = threadIdx.x >> 5, lane = threadIdx.x & 31;
  int row = blockIdx.x * 8 + wave;            // 16000
  int b = row / N_, n = row % N_;
  const float* xr = X + (size_t)(b*ROWS_ + n)*L_;
  float a0 = b1[lane], a1 = b1[lane+32], a2 = b1[lane+64];
  #pragma unroll
  for (int k = 0; k < L_; ++k) {
    float v = xr[k];
    a0 = fmaf(v, W1[k*T_ + lane],    a0);
    a1 = fmaf(v, W1[k*T_ + lane+32], a1);
    a2 = fmaf(v, W1[k*T_ + lane+64], a2);
  }
  hb[wave][lane] = elu_(a0); hb[wave][lane+32] = elu_(a1); hb[wave][lane+64] = elu_(a2);
  float o0 = b2[lane], o1 = b2[lane+32], o2 = b2[lane+64];
  for (int k = 0; k < T_; ++k) {
    float v = hb[wave][k];
    o0 = fmaf(v, W2[k*T_ + lane],    o0);
    o1 = fmaf(v, W2[k*T_ + lane+32], o1);
    o2 = fmaf(v, W2[k*T_ + lane+64], o2);
  }
  float* c = comb + (size_t)(b*N_ + n)*T_;
  c[lane] = o0; c[lane+32] = o1; c[lane+64] = o2;
}

// ---- 6) tr GRU input gates: GI2 = Xt2 @ tr_Wih^T + tr_bih  (B,T,6000) ----
__global__ void k_trgi(const float* __restrict__ X, const float* __restrict__ Wih,
                       const float* __restrict__ bih, float* __restrict__ GI2)
{
  size_t idx = (size_t)blockIdx.x * blockDim.x + threadIdx.x;   // B_*T_*G3_
  int o = (int)(idx % G3_);
  int bt = (int)(idx / G3_);
  int t = bt % T_, b = bt / T_;
  const float* xr = X + (size_t)(b*ROWS_ + N_ + t)*L_;
  const float* w = Wih + (size_t)o*L_;
  float a = bih[o];
  #pragma unroll
  for (int k = 0; k < L_; ++k) a = fmaf(xr[k], w[k], a);
  GI2[idx] = a;
}

// ---- 7) pack-transpose Whh(6000x2000) -> PW[k/2][o][2] so each WMMA B-frag
//         (two consecutive K values of one output column) is one b64 load ----
__global__ void k_tw(const float* __restrict__ Whh, float* __restrict__ PW)
{
  size_t idx = (size_t)blockIdx.x * blockDim.x + threadIdx.x;   // 12,000,000
  int o = (int)(idx / N_), k = (int)(idx % N_);
  PW[((size_t)(k >> 1)*G3_ + o)*2 + (k & 1)] = Whh[idx];
}

__global__ void k_zero(float* __restrict__ p, int n)
{ int i = blockIdx.x*blockDim.x + threadIdx.x; if (i < n) p[i] = 0.f; }

// ---- 8) tr GRU step t: gh = h @ Whh^T + bhh (WMMA, 3 gates), fused GRU update ----
__global__ void __launch_bounds__(256) k_trstep(const float* __restrict__ hIn,
    float* __restrict__ hOut, const float* __restrict__ PW,
    const float* __restrict__ bhh, const float* __restrict__ GI2,
    float* __restrict__ comb, int t)
{
  __shared__ float sP[3][8][128];     // gate, wave, (batch 0..7)*16 + col
  int wave = threadIdx.x >> 5, lane = threadIdx.x & 31;
  int m16 = lane & 15, off = (lane >> 4) << 1;
  int j0 = blockIdx.x * 16;
  v8f ar = vzero8(), az = vzero8(), an = vzero8();
  for (int ch = wave; ch < 500; ch += 8) {      // K=2000 in chunks of 4, split on waves
    int ks = ch*4 + off;                        // even
    v2f aH = (m16 < 8) ? *(const v2f*)(hIn + m16*N_ + ks) : vzero2();
    const float* wrow = PW + ((size_t)(ks >> 1)*G3_ + j0 + m16)*2;
    v2f b0 = *(const v2f*)(wrow);               // r gate
    v2f b1 = *(const v2f*)(wrow + (size_t)N_*2);     // z gate
    v2f b2 = *(const v2f*)(wrow + (size_t)2*N_*2);   // n gate
    ar = wmma4(aH, b0, ar);
    az = wmma4(aH, b1, az);
    an = wmma4(aH, b2, an);
  }
  if (lane < 16) {                     // lanes 0-15 hold valid rows 0..7
    #pragma unroll
    for (int i = 0; i < 8; ++i) {
      sP[0][wave][i*16 + m16] = ar[i];
      sP[1][wave][i*16 + m16] = az[i];
      sP[2][wave][i*16 + m16] = an[i];
    }
  }
  __syncthreads();
  if (threadIdx.x < 128) {             // 8 batches x 16 cols
    int bb = threadIdx.x >> 4, c = threadIdx.x & 15;
    int j = j0 + c;
    float gr = bhh[j], gz = bhh[N_ + j], gn = bhh[2*N_ + j];
    #pragma unroll
    for (int w = 0; w < 8; ++w) {
      gr += sP[0][w][bb*16 + c];
      gz += sP[1][w][bb*16 + c];
      gn += sP[2][w][bb*16 + c];
    }
    const float* gi = GI2 + (size_t)(bb*T_ + t)*G3_;
    float r  = sig_(gi[j] + gr);
    float z  = sig_(gi[N_ + j] + gz);
    float nn = tanhf(gi[2*N_ + j] + r*gn);
    float h  = (1.f - z)*nn + z*hIn[bb*N_ + j];
    hOut[bb*N_ + j] = h;
    comb[((size_t)bb*N_ + j)*T_ + t] += h;     // rs already there; += transposed rt
  }
}

// ---- 9) out = combined @ out_W + out_b  -> (B,N,1) ----
__global__ void k_out(const float* __restrict__ comb, const float* __restrict__ oW,
                      const float* __restrict__ ob, float* __restrict__ out)
{
  int idx = blockIdx.x*blockDim.x + threadIdx.x;   // 16000
  if (idx >= B_*N_) return;
  const float* c = comb + (size_t)idx*T_;
  float a = ob[0];
  #pragma unroll 8
  for (int t = 0; t < T_; ++t) a = fmaf(c[t], oW[t], a);
  out[idx] = a;
}

extern "C" void kernel_launch(void* const* d_in, const int* in_sizes, int n_in,
                              void* d_out, int out_size, void* d_ws, size_t ws_size,
                              hipStream_t stream)
{
  (void)in_sizes; (void)n_in; (void)out_size; (void)ws_size;
  const float* x      = (const float*)d_in[0];
  const float* x_mark = (const float*)d_in[1];
  const float* sp_emb = (const float*)d_in[2];
  const float* sp_W1  = (const float*)d_in[3];
  const float* sp_b1  = (const float*)d_in[4];
  const float* sp_W2  = (const float*)d_in[5];
  const float* sp_b2  = (const float*)d_in[6];
  const float* t_emb  = (const float*)d_in[7];
  const float* t_Wih  = (const float*)d_in[8];
  const float* t_Whh  = (const float*)d_in[9];
  const float* t_bih  = (const float*)d_in[10];
  const float* t_bhh  = (const float*)d_in[11];
  const float* gcn_W  = (const float*)d_in[12];
  const float* gcn_b  = (const float*)d_in[13];
  const float* fr_W1  = (const float*)d_in[14];
  const float* fr_b1  = (const float*)d_in[15];
  const float* fr_W2  = (const float*)d_in[16];
  const float* fr_b2  = (const float*)d_in[17];
  const float* tr_Wih = (const float*)d_in[18];
  const float* tr_Whh = (const float*)d_in[19];
  const float* tr_bih = (const float*)d_in[20];
  const float* tr_bhh = (const float*)d_in[21];
  const float* out_W  = (const float*)d_in[22];
  const float* out_b  = (const float*)d_in[23];

  // workspace layout (~80 MB of fp32)
  float* ws = (float*)d_ws;
  size_t o = 0;
  float* X0   = ws + o; o += (size_t)B_*ROWS_*L_;    // 536576
  float* X1   = ws + o; o += (size_t)B_*ROWS_*L_;
  float* X2   = ws + o; o += (size_t)B_*ROWS_*L_;
  float* GI   = ws + o; o += (size_t)B_*T_*96;
  float* GI2  = ws + o; o += (size_t)B_*T_*G3_;      // 4.6M
  float* PW   = ws + o; o += (size_t)N_*G3_;         // 12M (packed Whh^T)
  float* hA   = ws + o; o += (size_t)B_*N_;
  float* hB   = ws + o; o += (size_t)B_*N_;
  float* comb = ws + o; o += (size_t)B_*N_*T_;       // 1.5M

  k_spatial<<<2000, 256, 0, stream>>>(x, sp_emb, sp_W1, sp_b1, sp_W2, sp_b2, X0);
  k_tgi<<<96, 256, 0, stream>>>(x, x_mark, t_emb, t_Wih, t_bih, GI);
  k_tscan<<<1, 256, 0, stream>>>(GI, t_Whh, t_bhh, X0);

  dim3 gg(17, 8);   // 17*8 waves >= 131 row-tiles, per batch
  // TN layer 0 (A from X0, reused for both GCN layers via recompute)
  k_gcn<<<gg, 256, 0, stream>>>(X0, X0, X1, gcn_W + 0*1024, gcn_b + 0*32);
  k_gcn<<<gg, 256, 0, stream>>>(X0, X1, X2, gcn_W + 1*1024, gcn_b + 1*32);
  // TN layer 1 (A from X2)
  k_gcn<<<gg, 256, 0, stream>>>(X2, X2, X0, gcn_W + 2*1024, gcn_b + 2*32);
  k_gcn<<<gg, 256, 0, stream>>>(X2, X0, X1, gcn_W + 3*1024, gcn_b + 3*32);
  // final X in X1

  k_fr<<<2000, 256, 0, stream>>>(X1, fr_W1, fr_b1, fr_W2, fr_b2, comb);
  k_trgi<<<18000, 256, 0, stream>>>(X1, tr_Wih, tr_bih, GI2);
  k_tw<<<46875, 256, 0, stream>>>(tr_Whh, PW);
  k_zero<<<63, 256, 0, stream>>>(hA, B_*N_);

  for (int t = 0; t < T_; ++t) {
    float* hin  = (t & 1) ? hB : hA;
    float* hout = (t & 1) ? hA : hB;
    k_trstep<<<125, 256, 0, stream>>>(hin, hout, PW, tr_bhh, GI2, comb, t);
  }

  k_out<<<63, 256, 0, stream>>>(comb, out_W, out_b, (float*)d_out);
}